// AggregationEncoder_72773925863845
// MI455X (gfx1250) — compile-verified
//
#include <hip/hip_runtime.h>
#include <stdint.h>

#define NUM_GRID_NODES 262144
#define NUM_MESH_NODES 40962
#define EMBED          64
#define BATCH          4
#define PAD32          40992   // NUM_MESH_NODES rounded up to multiple of 32

// ---------------------------------------------------------------------------
// 1) Histogram of destination nodes (u32 atomics, cheap).
// ---------------------------------------------------------------------------
__global__ void count_kernel(const int2* __restrict__ ei,
                             unsigned* __restrict__ counts, int nedges) {
    int e = blockIdx.x * blockDim.x + threadIdx.x;
    if (e < nedges) {
        int2 sd = ei[e];                       // one global_load_b64
        atomicAdd(&counts[sd.y], 1u);
    }
}

// ---------------------------------------------------------------------------
// 2) Exclusive prefix sum of counts -> offsets[NUM_MESH_NODES+1].
//    Single 1024-thread block, chunked LDS scan with running carry.
// ---------------------------------------------------------------------------
__global__ void scan_kernel(const unsigned* __restrict__ counts,
                            unsigned* __restrict__ offsets, int n) {
    __shared__ unsigned buf[1024];
    __shared__ unsigned carry;
    int t = threadIdx.x;
    if (t == 0) carry = 0u;
    __syncthreads();
    for (int base = 0; base < n; base += 1024) {
        int i = base + t;
        unsigned v = (i < n) ? counts[i] : 0u;
        buf[t] = v;
        __syncthreads();
        for (int d = 1; d < 1024; d <<= 1) {
            unsigned add = (t >= d) ? buf[t - d] : 0u;
            __syncthreads();
            buf[t] += add;
            __syncthreads();
        }
        unsigned c = carry;                 // everyone reads old carry
        if (i < n) offsets[i] = c + buf[t] - v;
        __syncthreads();
        if (t == 0) carry = c + buf[1023];
        __syncthreads();
    }
    if (t == 0) offsets[n] = carry;
}

// ---------------------------------------------------------------------------
// 3) Bin src indices into CSR order by dst (u32 atomics on cursors).
// ---------------------------------------------------------------------------
__global__ void bin_kernel(const int2* __restrict__ ei,
                           const unsigned* __restrict__ offsets,
                           unsigned* __restrict__ cursor,
                           unsigned* __restrict__ edge_src, int nedges) {
    int e = blockIdx.x * blockDim.x + threadIdx.x;
    if (e < nedges) {
        int2 sd = ei[e];                       // one global_load_b64
        unsigned r = atomicAdd(&cursor[sd.y], 1u);
        edge_src[offsets[sd.y] + r] = (unsigned)sd.x;
    }
}

// ---------------------------------------------------------------------------
// 4) Gather-sum: one wave32 per mesh node, all 4 batches at once.
//    lane = batch*8 + slice ; each lane owns 8 consecutive floats (2x float4).
//    Edge-src chunks are staged into LDS via the CDNA5 async-to-LDS path,
//    then consumed 4 edges at a time (ds_load_b128 + 8 in-flight b128 loads).
// ---------------------------------------------------------------------------
__global__ void __launch_bounds__(256)
sum_kernel(const float* __restrict__ feat,
           const unsigned* __restrict__ edge_src,
           const unsigned* __restrict__ offsets,
           const unsigned* __restrict__ counts,
           float* __restrict__ out) {
    __shared__ __attribute__((aligned(16))) unsigned es[8 * 32]; // 32 u32/wave

    int gwave = (int)((blockIdx.x * blockDim.x + threadIdx.x) >> 5);
    int lane  = (int)(threadIdx.x & 31);
    if (gwave >= NUM_MESH_NODES) return;       // wave-uniform exit, no barriers
    int node = gwave;

    unsigned beg = offsets[node];
    unsigned end = offsets[node + 1];

    int b    = lane >> 3;                      // batch 0..3
    int part = lane & 7;                       // 8-float slice 0..7
    const float* fb = feat + ((size_t)b * NUM_GRID_NODES) * EMBED + part * 8;

    float4 acc0 = make_float4(0.f, 0.f, 0.f, 0.f);
    float4 acc1 = make_float4(0.f, 0.f, 0.f, 0.f);

    unsigned* myes = es + ((threadIdx.x >> 5) << 5);   // 128B slot, 16B aligned
    uint32_t lds_slot = (uint32_t)(uintptr_t)myes + (uint32_t)(lane << 2);

    for (unsigned c = beg; c < end; c += 32u) {
        // clamp so all 32 lanes issue a valid async load (extras are ignored)
        unsigned idx = c + (unsigned)lane;
        unsigned last = end - 1u;
        if (idx > last) idx = last;
        uint32_t voff = idx << 2;              // byte offset from edge_src base

        // drain pending LDS reads of the previous chunk, then async-stage the
        // next 32 src indices: each lane copies 4 bytes at edge_src + voff to LDS at lds_slot (GVS mode)
        asm volatile("s_wait_dscnt 0\n\t"
                     "global_load_async_to_lds_b32 %0, %1, %2"
                     :
                     : "v"(lds_slot), "v"(voff), "s"(edge_src)
                     : "memory");
        asm volatile("s_wait_asynccnt 0" ::: "memory");

        unsigned nchunk = end - c;
        if (nchunk > 32u) nchunk = 32u;

        unsigned j = 0;
        // 4 edges / iteration: 1x ds_load_b128 + 8 independent global_load_b128
        for (; j + 4u <= nchunk; j += 4u) {
            uint4 s4 = *(const uint4*)(myes + j);
            const float* p0 = fb + (size_t)s4.x * EMBED;
            const float* p1 = fb + (size_t)s4.y * EMBED;
            const float* p2 = fb + (size_t)s4.z * EMBED;
            const float* p3 = fb + (size_t)s4.w * EMBED;
            float4 x0 = ((const float4*)p0)[0], y0 = ((const float4*)p0)[1];
            float4 x1 = ((const float4*)p1)[0], y1 = ((const float4*)p1)[1];
            float4 x2 = ((const float4*)p2)[0], y2 = ((const float4*)p2)[1];
            float4 x3 = ((const float4*)p3)[0], y3 = ((const float4*)p3)[1];
            acc0.x += x0.x; acc0.y += x0.y; acc0.z += x0.z; acc0.w += x0.w;
            acc1.x += y0.x; acc1.y += y0.y; acc1.z += y0.z; acc1.w += y0.w;
            acc0.x += x1.x; acc0.y += x1.y; acc0.z += x1.z; acc0.w += x1.w;
            acc1.x += y1.x; acc1.y += y1.y; acc1.z += y1.z; acc1.w += y1.w;
            acc0.x += x2.x; acc0.y += x2.y; acc0.z += x2.z; acc0.w += x2.w;
            acc1.x += y2.x; acc1.y += y2.y; acc1.z += y2.z; acc1.w += y2.w;
            acc0.x += x3.x; acc0.y += x3.y; acc0.z += x3.z; acc0.w += x3.w;
            acc1.x += y3.x; acc1.y += y3.y; acc1.z += y3.z; acc1.w += y3.w;
        }
        for (; j < nchunk; ++j) {              // tail (<= 3 edges)
            unsigned src = myes[j];
            const float* p = fb + (size_t)src * EMBED;
            float4 x = ((const float4*)p)[0];
            float4 y = ((const float4*)p)[1];
            acc0.x += x.x; acc0.y += x.y; acc0.z += x.z; acc0.w += x.w;
            acc1.x += y.x; acc1.y += y.y; acc1.z += y.z; acc1.w += y.w;
        }
    }

    float cnt = (float)counts[node];
    float inv = 1.0f / fmaxf(cnt, 1.0f);
    float* op = out + ((size_t)b * NUM_MESH_NODES + (size_t)node) * EMBED + part * 8;
    float4 r0 = make_float4(acc0.x * inv, acc0.y * inv, acc0.z * inv, acc0.w * inv);
    float4 r1 = make_float4(acc1.x * inv, acc1.y * inv, acc1.z * inv, acc1.w * inv);
    ((float4*)op)[0] = r0;
    ((float4*)op)[1] = r1;
}

// ---------------------------------------------------------------------------
extern "C" void kernel_launch(void* const* d_in, const int* in_sizes, int n_in,
                              void* d_out, int out_size, void* d_ws, size_t ws_size,
                              hipStream_t stream) {
    const float* feat = (const float*)d_in[0];
    const int2*  ei   = (const int2*)d_in[1];
    float*       out  = (float*)d_out;

    int nedges = in_sizes[1] / 2;

    // workspace layout (u32): counts | cursor | offsets(+1) | edge_src
    unsigned* counts   = (unsigned*)d_ws;
    unsigned* cursor   = counts + PAD32;
    unsigned* offsets  = cursor + PAD32;
    unsigned* edge_src = offsets + PAD32;

    // zero counts + cursor (adjacent) — graph-capture-safe memset node
    hipMemsetAsync(d_ws, 0, (size_t)2 * PAD32 * sizeof(unsigned), stream);

    count_kernel<<<(nedges + 255) / 256, 256, 0, stream>>>(ei, counts, nedges);
    scan_kernel<<<1, 1024, 0, stream>>>(counts, offsets, NUM_MESH_NODES);
    bin_kernel<<<(nedges + 255) / 256, 256, 0, stream>>>(ei, offsets, cursor,
                                                         edge_src, nedges);

    int blocks = (NUM_MESH_NODES + 7) / 8;     // 8 waves (nodes) per 256-thread block
    sum_kernel<<<blocks, 256, 0, stream>>>(feat, edge_src, offsets, counts, out);
}